// SSREncoder_87505663689494
// MI455X (gfx1250) — compile-verified
//
#include <hip/hip_runtime.h>
#include <hip/hip_bf16.h>

#define DIM 128
#define TAU_INV 10.0f
#define EPSF 1e-10f
#define NEG_SLOPE 0.2f

typedef __attribute__((ext_vector_type(2))) float v2f;
typedef __attribute__((ext_vector_type(8))) float v8f;

// ---- helpers ---------------------------------------------------------------

// order-preserving float -> uint encoding (for atomicMax-based segment max)
__device__ __forceinline__ unsigned fenc(float f) {
    unsigned u = __float_as_uint(f);
    return (u & 0x80000000u) ? ~u : (u | 0x80000000u);
}
__device__ __forceinline__ float fdec(unsigned u) {
    u = (u & 0x80000000u) ? (u & 0x7fffffffu) : ~u;
    return __uint_as_float(u);
}
__device__ __forceinline__ float leaky(float v) { return v > 0.f ? v : NEG_SLOPE * v; }
__device__ __forceinline__ void atomAddF(float* p, float v) {
    __hip_atomic_fetch_add(p, v, __ATOMIC_RELAXED, __HIP_MEMORY_SCOPE_AGENT);
}

// ---- kernel 1: h = x @ W + b via v_wmma_f32_16x16x4_f32 --------------------
// block = 256 threads = 8 waves; block handles 16 node rows, each wave one
// 16-wide column tile. W (64KB) staged in LDS once per block.
__global__ __launch_bounds__(256) void gemm_h_kernel(
    const float* __restrict__ x, const float* __restrict__ W,
    const float* __restrict__ bias, float* __restrict__ h, int n_nodes)
{
    __shared__ float Wlds[DIM * DIM]; // 64 KB (WGP has 320 KB LDS)
    for (int i = threadIdx.x; i < DIM * DIM / 4; i += 256)
        ((float4*)Wlds)[i] = ((const float4*)W)[i];
    __syncthreads();

    const int wave    = threadIdx.x >> 5;
    const int lane    = threadIdx.x & 31;
    const int rowbase = blockIdx.x * 16;
    const int colbase = wave * 16;
    if (rowbase >= n_nodes) return;

    const int M  = lane & 15;          // A row / C column index within tile
    const int kb = (lane >> 4) << 1;   // lanes 0-15: K pair {0,1}; 16-31: {2,3}

    v8f c = {};
    const float* arow = x + (size_t)(rowbase + M) * DIM;
    for (int k = 0; k < DIM; k += 4) {
        // A fragment: lane holds A[M][k+kb], A[M][k+kb+1]
        float2 af = *(const float2*)(arow + k + kb);
        v2f a; a.x = af.x; a.y = af.y;
        // B fragment: lane holds W[k+kb][col], W[k+kb+1][col]
        v2f b;
        b.x = Wlds[(k + kb) * DIM + colbase + M];
        b.y = Wlds[(k + kb + 1) * DIM + colbase + M];
        c = __builtin_amdgcn_wmma_f32_16x16x4_f32(false, a, false, b,
                                                  (short)0, c, false, false);
    }
    // C/D layout: lanes 0-15 -> rows v, lanes 16-31 -> rows v+8; col = lane&15
    const float bN   = bias[colbase + M];
    const int mbase  = (lane < 16) ? 0 : 8;
    #pragma unroll
    for (int v = 0; v < 8; ++v)
        h[(size_t)(rowbase + mbase + v) * DIM + colbase + M] = c[v] + bN;
}

// ---- kernel 2: per-node attention scores s,t (wave per node) ---------------
__global__ __launch_bounds__(256) void score_kernel(
    const float* __restrict__ h, const float* __restrict__ att_src,
    const float* __restrict__ att_dst, float* __restrict__ s,
    float* __restrict__ t, int n_nodes)
{
    const int wave = threadIdx.x >> 5, lane = threadIdx.x & 31;
    const int node = blockIdx.x * 8 + wave;
    if (node >= n_nodes) return;
    const float4 hv = ((const float4*)(h + (size_t)node * DIM))[lane];
    const float4 as = ((const float4*)att_src)[lane];
    const float4 ad = ((const float4*)att_dst)[lane];
    float ps = hv.x * as.x + hv.y * as.y + hv.z * as.z + hv.w * as.w;
    float pt = hv.x * ad.x + hv.y * ad.y + hv.z * ad.z + hv.w * ad.w;
    #pragma unroll
    for (int off = 16; off > 0; off >>= 1) {
        ps += __shfl_xor(ps, off, 32);
        pt += __shfl_xor(pt, off, 32);
    }
    if (lane == 0) { s[node] = leaky(ps); t[node] = leaky(pt); }
}

// ---- kernel 3: per-hop init (m = enc(-FLT_MAX), z = 0, agg = 0) ------------
__global__ __launch_bounds__(256) void init_kernel(
    unsigned* __restrict__ m_enc, float* __restrict__ z,
    float* __restrict__ agg, int n_nodes, long long total)
{
    long long i = (long long)blockIdx.x * 256 + threadIdx.x;
    if (i < n_nodes) { m_enc[i] = 0x00800000u; z[i] = 0.f; } // enc(-FLT_MAX)
    if (i < total) agg[i] = 0.f;
}

// ---- kernel 4: edge logits + segment max -----------------------------------
__global__ __launch_bounds__(256) void logit_kernel(
    const int* __restrict__ src, const int* __restrict__ dst,
    const float* __restrict__ s, const float* __restrict__ t,
    const float* __restrict__ u, float* __restrict__ l_buf,
    unsigned* __restrict__ m_enc, int n_edges)
{
    int e = blockIdx.x * 256 + threadIdx.x;
    if (e >= n_edges) return;
    float g = -logf(-logf(u[e] + EPSF) + EPSF);
    float l = (s[src[e]] + t[dst[e]] + g) * TAU_INV;
    l_buf[e] = l;
    atomicMax(m_enc + dst[e], fenc(l));
}

// ---- kernel 5: exp(l - m[dst]) + segment sum -------------------------------
__global__ __launch_bounds__(256) void exp_kernel(
    const int* __restrict__ dst, const unsigned* __restrict__ m_enc,
    float* __restrict__ l_buf, float* __restrict__ z, int n_edges)
{
    int e = blockIdx.x * 256 + threadIdx.x;
    if (e >= n_edges) return;
    int d = dst[e];
    float ex = expf(l_buf[e] - fdec(m_enc[d]));
    l_buf[e] = ex;                    // reuse buffer in place
    atomAddF(z + d, ex);
}

// ---- kernel 6: alpha * h[src] scatter-add (wave per edge) ------------------
__global__ __launch_bounds__(256) void scatter_kernel(
    const int* __restrict__ src, const int* __restrict__ dst,
    const float* __restrict__ e_buf, const float* __restrict__ z,
    const float* __restrict__ h, float* __restrict__ agg, int n_edges)
{
    const int wave = threadIdx.x >> 5, lane = threadIdx.x & 31;
    long long edge = (long long)blockIdx.x * 8 + wave;
    if (edge >= n_edges) return;
    const int sN = src[edge], dN = dst[edge];
    const float alpha = e_buf[edge] / (z[dN] + EPSF);
    const float4 hv = ((const float4*)(h + (size_t)sN * DIM))[lane];
    float* ap = agg + (size_t)dN * DIM + lane * 4;
    atomAddF(ap + 0, alpha * hv.x);
    atomAddF(ap + 1, alpha * hv.y);
    atomAddF(ap + 2, alpha * hv.z);
    atomAddF(ap + 3, alpha * hv.w);
}

// ---- kernel 7: residual + relu ---------------------------------------------
__global__ __launch_bounds__(256) void residual_kernel(
    const float* __restrict__ agg, const float* __restrict__ xin,
    float* __restrict__ xout, long long total)
{
    long long i = (long long)blockIdx.x * 256 + threadIdx.x;
    if (i >= total) return;
    float v = agg[i] + xin[i];
    xout[i] = v > 0.f ? v : 0.f;
}

// ---- kernel 8: gather roots ------------------------------------------------
__global__ __launch_bounds__(256) void gather_kernel(
    const float* __restrict__ xf, const int* __restrict__ roots,
    float* __restrict__ out, int n_roots)
{
    int i = blockIdx.x * 256 + threadIdx.x;
    if (i >= n_roots * DIM) return;
    int r = i / DIM, d = i - r * DIM;
    out[i] = xf[(size_t)roots[r] * DIM + d];
}

// ---- host launcher ---------------------------------------------------------
extern "C" void kernel_launch(void* const* d_in, const int* in_sizes, int n_in,
                              void* d_out, int out_size, void* d_ws, size_t ws_size,
                              hipStream_t stream) {
    const float* x        = (const float*)d_in[0];
    const int*   ei       = (const int*)d_in[1];
    const int*   roots    = (const int*)d_in[2];
    const float* gu       = (const float*)d_in[3];
    const float* W        = (const float*)d_in[4];
    const float* bias     = (const float*)d_in[5];
    const float* att_src  = (const float*)d_in[6];
    const float* att_dst  = (const float*)d_in[7];
    float*       out      = (float*)d_out;

    const int N = in_sizes[0] / DIM;       // 100000
    const int E = in_sizes[1] / 2;         // 1600000
    const int NR = in_sizes[2];            // 1024
    const long long total = (long long)N * DIM;

    const int* src = ei;
    const int* dst = ei + E;

    // workspace layout (f32 unless noted)
    float*    h    = (float*)d_ws;         // N*DIM
    float*    X1   = h + total;            // N*DIM
    float*    agg  = X1 + total;           // N*DIM
    float*    sarr = agg + total;          // N
    float*    tarr = sarr + N;             // N
    float*    z    = tarr + N;             // N
    unsigned* menc = (unsigned*)(z + N);   // N (uint)
    float*    lbuf = (float*)(menc + N);   // E

    const int gemm_blocks  = (N + 15) / 16;
    const int score_blocks = (N + 7) / 8;
    const int node_blocks  = (int)((total + 255) / 256);
    const int edge_blocks  = (E + 255) / 256;
    const int scat_blocks  = (E + 7) / 8;

    const float* xin = x;
    for (int hop = 0; hop < 2; ++hop) {
        gemm_h_kernel<<<gemm_blocks, 256, 0, stream>>>(xin, W, bias, h, N);
        score_kernel<<<score_blocks, 256, 0, stream>>>(h, att_src, att_dst, sarr, tarr, N);
        init_kernel<<<node_blocks, 256, 0, stream>>>(menc, z, agg, N, total);
        logit_kernel<<<edge_blocks, 256, 0, stream>>>(src, dst, sarr, tarr,
                                                      gu + (size_t)hop * E, lbuf, menc, E);
        exp_kernel<<<edge_blocks, 256, 0, stream>>>(dst, menc, lbuf, z, E);
        scatter_kernel<<<scat_blocks, 256, 0, stream>>>(src, dst, lbuf, z, h, agg, E);
        residual_kernel<<<node_blocks, 256, 0, stream>>>(agg, xin, X1, total);
        xin = X1;
    }
    gather_kernel<<<(NR * DIM + 255) / 256, 256, 0, stream>>>(X1, roots, out, NR);
}